// MSCAM_a_67912022885014
// MI455X (gfx1250) — compile-verified
//
#include <hip/hip_runtime.h>
#include <hip/hip_bf16.h>

typedef __attribute__((ext_vector_type(2))) float v2f;
typedef __attribute__((ext_vector_type(8))) float v8f;

#define CIN   256
#define HW    4096
#define TFR   8
#define NB    8
#define CFEAT 256
#define CATT  64
#define CTX_ELEMS (NB * CFEAT * (TFR * CATT))   // 8*256*512 = 1,048,576

// D = W(Cout x 256) @ X(256 x 4096) + bias, one frame per blockIdx.z.
// 8 waves per block: wave = (mw 0..3) x (nw 0..1); each wave -> 16 rows x 64 cols
// as 4 independent 16x16 WMMA accumulators (ILP over the WMMA RAW hazard).
__global__ __launch_bounds__(256) void conv1x1_wmma(
    const float* __restrict__ X, const float* __restrict__ W,
    const float* __restrict__ Bias, float* __restrict__ Out,
    int Cout, int frame_step)
{
    const size_t frame_elems = (size_t)CIN * HW;
    const float* __restrict__ Xf = X + (size_t)blockIdx.z * frame_step * frame_elems;
    float* __restrict__ Of = Out + (size_t)blockIdx.z * (size_t)Cout * HW;

    const int tid  = threadIdx.x;
    const int wave = tid >> 5;
    const int lane = tid & 31;
    const int lm   = lane & 15;        // N (for B/C/D) or M (for A)
    const int kb   = (lane >> 4) << 1; // 0 or 2: K-half select for A/B fragments

    const int m0    = blockIdx.y * 64 + (wave & 3) * 16;
    const int nbase = blockIdx.x * 128 + (wave >> 2) * 64;

    v8f acc0 = {}, acc1 = {}, acc2 = {}, acc3 = {};

    const float* __restrict__ Wrow = W + (size_t)(m0 + lm) * CIN;

    for (int c = 0; c < CIN; c += 4) {
        // A fragment: 16x4 chunk of W. lane lm = row M, kb selects K pair.
        v2f a = *(const v2f*)(Wrow + c + kb);

        // prefetch the X rows 4 k-steps ahead (speculative; OOB prefetch is dropped)
        __builtin_prefetch(Xf + (size_t)(c + 16 + kb) * HW + nbase + lane, 0, 1);

        // B fragments: 4x16 chunks of X. lane lm = col N, kb selects K pair.
        const float* __restrict__ Xr0 = Xf + (size_t)(c + kb) * HW + lm;
        const float* __restrict__ Xr1 = Xr0 + HW;

        v2f b0, b1, b2, b3;
        b0.x = Xr0[nbase +  0]; b0.y = Xr1[nbase +  0];
        b1.x = Xr0[nbase + 16]; b1.y = Xr1[nbase + 16];
        b2.x = Xr0[nbase + 32]; b2.y = Xr1[nbase + 32];
        b3.x = Xr0[nbase + 48]; b3.y = Xr1[nbase + 48];

        acc0 = __builtin_amdgcn_wmma_f32_16x16x4_f32(false, a, false, b0, (short)0, acc0, false, false);
        acc1 = __builtin_amdgcn_wmma_f32_16x16x4_f32(false, a, false, b1, (short)0, acc1, false, false);
        acc2 = __builtin_amdgcn_wmma_f32_16x16x4_f32(false, a, false, b2, (short)0, acc2, false, false);
        acc3 = __builtin_amdgcn_wmma_f32_16x16x4_f32(false, a, false, b3, (short)0, acc3, false, false);
    }

    // C/D layout: lane lm = column, VGPR r -> row m0 + r + 8*(lane>=16)
    const int mh = (lane >> 4) << 3;
#pragma unroll
    for (int r = 0; r < 8; ++r) {
        const int m = m0 + r + mh;
        const float bv = Bias[m];
        float* __restrict__ Orow = Of + (size_t)m * HW + lm;
        Orow[nbase +  0] = acc0[r] + bv;
        Orow[nbase + 16] = acc1[r] + bv;
        Orow[nbase + 32] = acc2[r] + bv;
        Orow[nbase + 48] = acc3[r] + bv;
    }
}

// In-place softmax over one 4096-element row per block (256 threads, 16 elems/thread).
__global__ __launch_bounds__(256) void softmax_rows(float* __restrict__ data)
{
    __shared__ float red[256];
    float* __restrict__ row = data + (size_t)blockIdx.x * HW;
    const int t = threadIdx.x;

    float v[16];
    float m = -3.402823466e38f;
#pragma unroll
    for (int i = 0; i < 16; ++i) {
        v[i] = row[t + i * 256];
        m = fmaxf(m, v[i]);
    }
    red[t] = m;
    __syncthreads();
    for (int s = 128; s > 0; s >>= 1) {
        if (t < s) red[t] = fmaxf(red[t], red[t + s]);
        __syncthreads();
    }
    const float rowmax = red[0];
    __syncthreads();

    float sum = 0.f;
#pragma unroll
    for (int i = 0; i < 16; ++i) {
        v[i] = __expf(v[i] - rowmax);
        sum += v[i];
    }
    red[t] = sum;
    __syncthreads();
    for (int s = 128; s > 0; s >>= 1) {
        if (t < s) red[t] += red[t + s];
        __syncthreads();
    }
    const float inv = 1.0f / red[0];
#pragma unroll
    for (int i = 0; i < 16; ++i)
        row[t + i * 256] = v[i] * inv;
}

// context[b][m][n] = sum_p feat[b][m][p] * atten[b][n][p]   (M=256, N=512, K=4096)
// 256-thread blocks, 64x64 C-tile per WG: 8 waves = (mw 0..3) x (nw 0..1),
// each wave a 16x32 strip = 2 N-subtiles x 2 K-interleaved accumulator chains
// (4 independent WMMA chains). Cuts unique L2 reads per WG to 2 MB
// ((64+64) rows x 4096 x 4B) -> ~512 MB total vs ~2 GB for 1 tile/wave.
__global__ __launch_bounds__(256) void context_wmma(
    const float* __restrict__ Feat, const float* __restrict__ Atten,
    float* __restrict__ Ctx)
{
    const int batch = blockIdx.z;
    const int wave  = threadIdx.x >> 5;
    const int lane  = threadIdx.x & 31;
    const int lm    = lane & 15;
    const int kb    = (lane >> 4) << 1;

    const int m0 = blockIdx.y * 64 + (wave & 3) * 16;       // 256/64 = 4 M-tiles
    const int n0 = blockIdx.x * 64 + (wave >> 2) * 32;      // 512/64 = 8 N-tiles

    const float* __restrict__ Arow  = Feat  + ((size_t)batch * CFEAT + m0 + lm) * HW + kb;
    const float* __restrict__ Brow0 = Atten + ((size_t)batch * (TFR * CATT) + n0 + lm) * HW + kb;
    const float* __restrict__ Brow1 = Brow0 + (size_t)16 * HW;

    v8f acc00 = {}, acc01 = {}, acc10 = {}, acc11 = {};

    for (int k = 0; k < HW; k += 8) {
        v2f a0  = *(const v2f*)(Arow  + k);
        v2f a1  = *(const v2f*)(Arow  + k + 4);
        v2f b00 = *(const v2f*)(Brow0 + k);
        v2f b01 = *(const v2f*)(Brow0 + k + 4);
        v2f b10 = *(const v2f*)(Brow1 + k);
        v2f b11 = *(const v2f*)(Brow1 + k + 4);

        acc00 = __builtin_amdgcn_wmma_f32_16x16x4_f32(false, a0, false, b00, (short)0, acc00, false, false);
        acc10 = __builtin_amdgcn_wmma_f32_16x16x4_f32(false, a0, false, b10, (short)0, acc10, false, false);
        acc01 = __builtin_amdgcn_wmma_f32_16x16x4_f32(false, a1, false, b01, (short)0, acc01, false, false);
        acc11 = __builtin_amdgcn_wmma_f32_16x16x4_f32(false, a1, false, b11, (short)0, acc11, false, false);
    }

    const int mh = (lane >> 4) << 3;
#pragma unroll
    for (int r = 0; r < 8; ++r) {
        const int m = m0 + r + mh;
        float* __restrict__ Crow = Ctx + ((size_t)batch * CFEAT + m) * (TFR * CATT);
        Crow[n0 + lm]      = acc00[r] + acc01[r];
        Crow[n0 + 16 + lm] = acc10[r] + acc11[r];
    }
}

extern "C" void kernel_launch(void* const* d_in, const int* in_sizes, int n_in,
                              void* d_out, int out_size, void* d_ws, size_t ws_size,
                              hipStream_t stream) {
    (void)in_sizes; (void)n_in; (void)out_size; (void)ws_size;
    const float* x       = (const float*)d_in[0];
    const float* W_feat  = (const float*)d_in[1];
    const float* b_feat  = (const float*)d_in[2];
    const float* W_atten = (const float*)d_in[3];
    const float* b_atten = (const float*)d_in[4];
    // d_in[5] is T (==8), baked into the constants above.

    float* ctx   = (float*)d_out;                 // (8, 256, 512)
    float* atten = (float*)d_out + CTX_ELEMS;     // (8, 512, 4096) == (64, 64, 4096)
    float* feat  = (float*)d_ws;                  // (8, 256, 4096) = 32 MB scratch

    // feat: 8 batches, frame = b*8, Cout=256  -> grid (N/128, Cout/64, 8)
    conv1x1_wmma<<<dim3(32, 4, NB), dim3(256), 0, stream>>>(x, W_feat, b_feat, feat, CFEAT, TFR);
    // atten_raw: 64 frames, Cout=64 -> grid (32, 1, 64), written straight into d_out
    conv1x1_wmma<<<dim3(32, 1, NB * TFR), dim3(256), 0, stream>>>(x, W_atten, b_atten, atten, CATT, 1);
    // softmax over 4096 rows of 4096, in place
    softmax_rows<<<4096, 256, 0, stream>>>(atten);
    // context: 8 batches x 4 M-tiles x 8 N-tiles, 64x64 per WG
    context_wmma<<<dim3(8, 4, NB), dim3(256), 0, stream>>>(feat, atten, ctx);
}